// NewDecoder_24507083391031
// MI455X (gfx1250) — compile-verified
//
#include <hip/hip_runtime.h>
#include <math.h>

// ---------------------------------------------------------------------------
// NewDecoder step for MI455X (gfx1250, wave32).
// FP32 WMMA 16x16x4 for all GEMMs (memory-bound problem: ~240MB traffic vs
// ~3.3 GFLOP; keep f32, split M=32 into two 16-row WMMA tiles).
// k_out (131MB W_out stream) uses gfx1250 async global->LDS staging
// (GLOBAL_LOAD_ASYNC_TO_LDS_B128 + s_wait_asynccnt) to feed WMMA from LDS.
// ---------------------------------------------------------------------------

typedef __attribute__((ext_vector_type(2))) float v2f;
typedef __attribute__((ext_vector_type(8))) float v8f;

#define Bsz 32
#define Hd  1024
#define Ld  128
#define Td  127
#define Vd  32000
#define NT  255          // L + T
#define KC  32           // k-chunk for LDS-staged GEMM

__device__ __forceinline__ v8f wmma4(v2f a, v2f b, v8f c) {
  // D = A(16x4 f32) * B(4x16 f32) + C(16x16 f32)
  return __builtin_amdgcn_wmma_f32_16x16x4_f32(
      /*neg_a=*/false, a, /*neg_b=*/false, b,
      /*c_mod=*/(short)0, c, /*reuse_a=*/false, /*reuse_b=*/false);
}

__device__ __forceinline__ v2f ld2(const float* p) {
  return *(const v2f*)p;   // callers guarantee 8B alignment (even k offsets)
}

__device__ __forceinline__ float sigm(float x) {
  return 1.0f / (1.0f + expf(-x));
}

// gfx1250 async copy: 16B per lane, global -> LDS, tracked on ASYNCcnt.
// lds_byte: 32-bit LDS byte address (generic shared ptr truncated; LDS
// aperture keeps the offset in addr[31:0]).
__device__ __forceinline__ void async_g2l_b128(unsigned lds_byte,
                                               const float* gptr) {
  asm volatile("global_load_async_to_lds_b128 %0, %1, off"
               :: "v"(lds_byte), "v"(gptr)
               : "memory");
}
__device__ __forceinline__ void wait_async0() {
  asm volatile("s_wait_asynccnt 0" ::: "memory");
}
__device__ __forceinline__ unsigned lds_addr32(const void* p) {
  return (unsigned)(unsigned long long)p;
}

// ---------------------------------------------------------------------------
// Kernel 1: x = emb[word_indices], h0 = last_hidden + tanh_hidden, begin flag
// ---------------------------------------------------------------------------
__global__ void __launch_bounds__(256) k_prep(
    const int* __restrict__ widx, const float* __restrict__ lhid,
    const float* __restrict__ thid, const float* __restrict__ emb,
    float* __restrict__ x, float* __restrict__ h0, int* __restrict__ flag) {
  int i = blockIdx.x * 256 + threadIdx.x;
  if (i < Bsz * Hd) {
    int b = i >> 10;
    int h = i & (Hd - 1);
    x[i]  = emb[(size_t)widx[b] * Hd + h];
    h0[i] = lhid[i] + thid[i];
  }
  if (blockIdx.x == 0 && threadIdx.x == 0) {
    int s = 0;
    for (int b = 0; b < Bsz; ++b) s += widx[b];
    *flag = (s == 0) ? 1 : 0;
  }
}

// ---------------------------------------------------------------------------
// Kernel 2: LSTM gates GEMM (WMMA f32) + fused activations + begin-select.
//   gates = x*W_ih^T + h0*W_hh^T + b_ih + b_hh ; c0 == 0 so f-gate drops out:
//   c1 = sigmoid(i)*tanh(g) ; h1 = sigmoid(o)*tanh(c1)
// One wave per 16-column tile of H (64 tiles). Grid 8 x 256 threads (8 waves).
// ---------------------------------------------------------------------------
__global__ void __launch_bounds__(256) k_lstm(
    const float* __restrict__ x, const float* __restrict__ h0,
    const float* __restrict__ Wih, const float* __restrict__ Whh,
    const float* __restrict__ bih, const float* __restrict__ bhh,
    const float* __restrict__ lhid, const int* __restrict__ flag,
    float* __restrict__ cur) {
  const int wid   = threadIdx.x >> 5;
  const int lane  = threadIdx.x & 31;
  const int ct    = blockIdx.x * 8 + wid;       // 0..63 column tile over H
  const int mrow  = lane & 15;                  // A-matrix M row (per layout)
  const int khalf = lane >> 4;                  // lanes 16-31 carry K=2,3
  const int ci    = ct * 16 + (lane & 15);      // column within H (B-tile N)

  const float* wI = Wih + (size_t)ci * Hd;              // i-gate rows
  const float* wG = Wih + (size_t)(2 * Hd + ci) * Hd;   // g-gate rows
  const float* wO = Wih + (size_t)(3 * Hd + ci) * Hd;   // o-gate rows
  const float* uI = Whh + (size_t)ci * Hd;
  const float* uG = Whh + (size_t)(2 * Hd + ci) * Hd;
  const float* uO = Whh + (size_t)(3 * Hd + ci) * Hd;

  v8f iL = {}, iH = {}, gL = {}, gH = {}, oL = {}, oH = {};

  for (int k = 0; k < Hd; k += 4) {
    const int ko = k + 2 * khalf;               // even -> 8B aligned
    // A tiles (rows 0-15 and 16-31) for both x and h0
    v2f axL = ld2(x  + mrow * Hd + ko);
    v2f axH = ld2(x  + (16 + mrow) * Hd + ko);
    v2f ahL = ld2(h0 + mrow * Hd + ko);
    v2f ahH = ld2(h0 + (16 + mrow) * Hd + ko);
    // B tiles: B[k][n] = W[n][k]
    v2f bI = ld2(wI + ko), bG = ld2(wG + ko), bO = ld2(wO + ko);
    v2f cI = ld2(uI + ko), cG = ld2(uG + ko), cO = ld2(uO + ko);

    iL = wmma4(axL, bI, iL);  iH = wmma4(axH, bI, iH);
    gL = wmma4(axL, bG, gL);  gH = wmma4(axH, bG, gH);
    oL = wmma4(axL, bO, oL);  oH = wmma4(axH, bO, oH);
    iL = wmma4(ahL, cI, iL);  iH = wmma4(ahH, cI, iH);
    gL = wmma4(ahL, cG, gL);  gH = wmma4(ahH, cG, gH);
    oL = wmma4(ahL, cO, oL);  oH = wmma4(ahH, cO, oH);
  }

  const int   isb = *flag;
  const float bI_ = bih[ci]          + bhh[ci];
  const float bG_ = bih[2 * Hd + ci] + bhh[2 * Hd + ci];
  const float bO_ = bih[3 * Hd + ci] + bhh[3 * Hd + ci];

  // D layout: VGPR v -> row v (lanes 0-15) / row v+8 (lanes 16-31), N=lane&15
  for (int v = 0; v < 8; ++v) {
    {
      const int r = v + 8 * khalf;
      float c1 = sigm(iL[v] + bI_) * tanhf(gL[v] + bG_);
      float h1 = sigm(oL[v] + bO_) * tanhf(c1);
      cur[r * Hd + ci] = isb ? lhid[r * Hd + ci] : h1;
    }
    {
      const int r = 16 + v + 8 * khalf;
      float c1 = sigm(iH[v] + bI_) * tanhf(gH[v] + bG_);
      float h1 = sigm(oH[v] + bO_) * tanhf(c1);
      cur[r * Hd + ci] = isb ? lhid[r * Hd + ci] : h1;
    }
  }
}

// ---------------------------------------------------------------------------
// Kernel 3: ragged attention. One block per batch element.
//   enc[t] = seq[t] (t<L) else tree[t-L]; masked softmax; context = w @ enc
// ---------------------------------------------------------------------------
__global__ void __launch_bounds__(256) k_attn(
    const float* __restrict__ cur, const float* __restrict__ seq,
    const float* __restrict__ tree, const int* __restrict__ numNode,
    float* __restrict__ ctx) {
  const int b    = blockIdx.x;
  const int tid  = threadIdx.x;
  const int wid  = tid >> 5;
  const int lane = tid & 31;

  __shared__ float sc[NT + 1];
  __shared__ float red[256];

  const float* curb = cur + b * Hd;

  // scores: one wave per position (strided)
  for (int t = wid; t < NT; t += 8) {
    const float* e = (t < Ld) ? (seq  + ((size_t)t * Bsz + b) * Hd)
                              : (tree + ((size_t)(t - Ld) * Bsz + b) * Hd);
    float p = 0.0f;
    for (int h = lane; h < Hd; h += 32) p += curb[h] * e[h];
    for (int off = 16; off; off >>= 1) p += __shfl_xor(p, off, 32);
    if (lane == 0) sc[t] = p;
  }
  if (tid == 0) sc[NT] = -1e30f;
  __syncthreads();

  // mask
  const int nn = numNode[b];
  if (tid < NT) {
    const bool valid = (tid < Ld) ? (tid < nn + 1) : ((tid - Ld) < nn);
    if (!valid) sc[tid] = -1e30f;
  }
  __syncthreads();

  // max-reduce
  red[tid] = sc[tid];
  __syncthreads();
  for (int s = 128; s; s >>= 1) {
    if (tid < s) red[tid] = fmaxf(red[tid], red[tid + s]);
    __syncthreads();
  }
  const float mx = red[0];
  __syncthreads();

  // exp + sum-reduce
  const float e = (tid < NT) ? expf(sc[tid] - mx) : 0.0f;
  red[tid] = e;
  __syncthreads();
  for (int s = 128; s; s >>= 1) {
    if (tid < s) red[tid] += red[tid + s];
    __syncthreads();
  }
  const float inv = 1.0f / red[0];
  __syncthreads();
  sc[tid] = e * inv;               // sc[255] = 0
  __syncthreads();

  // context (coalesced over h)
  for (int h = tid; h < Hd; h += 256) {
    float acc = 0.0f;
    for (int t = 0; t < NT; ++t) {
      const float* ep = (t < Ld) ? (seq  + ((size_t)t * Bsz + b) * Hd)
                                 : (tree + ((size_t)(t - Ld) * Bsz + b) * Hd);
      acc += sc[t] * ep[h];
    }
    ctx[b * Hd + h] = acc;
  }
}

// ---------------------------------------------------------------------------
// Kernel 4: tanh_h = tanh([cur, ctx] @ W_comb^T + b_comb)  (WMMA f32, K=2048)
// ---------------------------------------------------------------------------
__global__ void __launch_bounds__(256) k_comb(
    const float* __restrict__ cur, const float* __restrict__ ctx,
    const float* __restrict__ Wc, const float* __restrict__ bc,
    float* __restrict__ th) {
  const int wid   = threadIdx.x >> 5;
  const int lane  = threadIdx.x & 31;
  const int ct    = blockIdx.x * 8 + wid;       // 0..63
  const int mrow  = lane & 15;
  const int khalf = lane >> 4;
  const int col   = ct * 16 + (lane & 15);

  const float* wr = Wc + (size_t)col * (2 * Hd);
  v8f aL = {}, aH = {};

  for (int k = 0; k < Hd; k += 4) {
    const int ko = k + 2 * khalf;
    v2f xL = ld2(cur + mrow * Hd + ko);
    v2f xH = ld2(cur + (16 + mrow) * Hd + ko);
    v2f b1 = ld2(wr + ko);
    aL = wmma4(xL, b1, aL);
    aH = wmma4(xH, b1, aH);
    v2f yL = ld2(ctx + mrow * Hd + ko);
    v2f yH = ld2(ctx + (16 + mrow) * Hd + ko);
    v2f b2 = ld2(wr + Hd + ko);
    aL = wmma4(yL, b2, aL);
    aH = wmma4(yH, b2, aH);
  }

  const float bb = bc[col];
  for (int v = 0; v < 8; ++v) {
    const int r = v + 8 * khalf;
    th[r * Hd + col]        = tanhf(aL[v] + bb);
    th[(16 + r) * Hd + col] = tanhf(aH[v] + bb);
  }
}

// ---------------------------------------------------------------------------
// Kernel 5: out_vec = tanh_h @ W_out^T + b_out  (32 x 32000, K=1024)
// gfx1250 async global->LDS staging (ASYNCcnt) of W/A tiles, WMMA from LDS.
// Block: 256 thr (8 waves), 128 output cols per block, grid = 250.
// ---------------------------------------------------------------------------
__global__ void __launch_bounds__(256) k_out(
    const float* __restrict__ A, const float* __restrict__ W,
    const float* __restrict__ bias, float* __restrict__ out) {
  __shared__ float As[Bsz * KC];   // 32 x 32 activations
  __shared__ float Bs[128 * KC];   // 128 x 32 weight tile

  const int tid   = threadIdx.x;
  const int wid   = tid >> 5;
  const int lane  = tid & 31;
  const int col0  = blockIdx.x * 128;
  const int mrow  = lane & 15;
  const int khalf = lane >> 4;
  const int nloc  = wid * 16 + (lane & 15);     // local column 0..127

  const int r8 = tid >> 3;                      // 0..31 staging row
  const int k4 = (tid & 7) * 4;                 // 0,4,...,28 staging k

  const unsigned ldsA  = lds_addr32(&As[r8 * KC + k4]);
  const unsigned ldsB0 = lds_addr32(&Bs[0]);

  v8f accL = {}, accH = {};

  for (int kk = 0; kk < Hd; kk += KC) {
    // stage activations (32x32): 16B per lane, async into LDS
    async_g2l_b128(ldsA, A + (size_t)r8 * Hd + kk + k4);
    // stage weights (128x32), 4 passes of 32 rows; prefetch next chunk
    for (int p = 0; p < 4; ++p) {
      const int r = p * 32 + r8;
      const float* gw = W + (size_t)(col0 + r) * Hd + kk + k4;
      async_g2l_b128(ldsB0 + (unsigned)((r * KC + k4) * sizeof(float)), gw);
      if (kk + KC < Hd)
        __builtin_prefetch(gw + KC, 0, 1);      // global_prefetch_b8
    }
    wait_async0();                              // s_wait_asynccnt 0
    __syncthreads();

    for (int k2 = 0; k2 < KC; k2 += 4) {
      const int ko = k2 + 2 * khalf;
      v2f a0 = ld2(&As[mrow * KC + ko]);
      v2f a1 = ld2(&As[(16 + mrow) * KC + ko]);
      v2f bv = ld2(&Bs[nloc * KC + ko]);
      accL = wmma4(a0, bv, accL);
      accH = wmma4(a1, bv, accH);
    }
    __syncthreads();
  }

  const int   col = col0 + nloc;
  const float bb  = bias[col];
  for (int v = 0; v < 8; ++v) {
    const int r = v + 8 * khalf;
    out[(size_t)r * Vd + col]        = accL[v] + bb;
    out[(size_t)(16 + r) * Vd + col] = accH[v] + bb;
  }
}

// ---------------------------------------------------------------------------
// Kernel 6: softmax over dim 0 (batch), in place. One thread per vocab column.
// ---------------------------------------------------------------------------
__global__ void __launch_bounds__(256) k_smax0(float* __restrict__ p) {
  const int v = blockIdx.x * 256 + threadIdx.x;
  if (v >= Vd) return;
  float m = -3.4e38f;
  for (int b = 0; b < Bsz; ++b) m = fmaxf(m, p[(size_t)b * Vd + v]);
  float s = 0.0f;
  for (int b = 0; b < Bsz; ++b) s += expf(p[(size_t)b * Vd + v] - m);
  const float inv = 1.0f / s;
  for (int b = 0; b < Bsz; ++b)
    p[(size_t)b * Vd + v] = expf(p[(size_t)b * Vd + v] - m) * inv;
}

// ---------------------------------------------------------------------------
extern "C" void kernel_launch(void* const* d_in, const int* in_sizes, int n_in,
                              void* d_out, int out_size, void* d_ws,
                              size_t ws_size, hipStream_t stream) {
  const int*   widx = (const int*)d_in[0];
  const float* lhid = (const float*)d_in[1];   // (1,B,H) -> (B,H)
  const float* thid = (const float*)d_in[2];
  const float* tree = (const float*)d_in[3];   // (T,B,H)
  const float* seq  = (const float*)d_in[4];   // (L,B,H)
  const int*   nNod = (const int*)d_in[5];
  const float* emb  = (const float*)d_in[6];
  const float* Wih  = (const float*)d_in[7];
  const float* Whh  = (const float*)d_in[8];
  const float* bih  = (const float*)d_in[9];
  const float* bhh  = (const float*)d_in[10];
  const float* Wc   = (const float*)d_in[11];
  const float* bc   = (const float*)d_in[12];
  const float* Wo   = (const float*)d_in[13];
  const float* bo   = (const float*)d_in[14];

  float* out  = (float*)d_out;
  float* prob = out;                       // (B,V)  1,024,000
  float* cur  = out + Bsz * Vd;            // (B,H)
  float* th   = cur + Bsz * Hd;            // (B,H)
  float* ctx  = th + Bsz * Hd;             // (B,H)

  float* x  = (float*)d_ws;                // (B,H) embedded words
  float* h0 = x + Bsz * Hd;                // (B,H) lhidden + tanh_hidden
  int* flag = (int*)(h0 + Bsz * Hd);       // is_begin

  k_prep<<<(Bsz * Hd + 255) / 256, 256, 0, stream>>>(widx, lhid, thid, emb, x,
                                                     h0, flag);
  k_lstm<<<8, 256, 0, stream>>>(x, h0, Wih, Whh, bih, bhh, lhid, flag, cur);
  k_attn<<<Bsz, 256, 0, stream>>>(cur, seq, tree, nNod, ctx);
  k_comb<<<8, 256, 0, stream>>>(cur, ctx, Wc, bc, th);
  k_out<<<Vd / 128, 256, 0, stream>>>(th, Wo, bo, prob);
  k_smax0<<<(Vd + 255) / 256, 256, 0, stream>>>(prob);
}